// PhiQComputer_47656957116943
// MI455X (gfx1250) — compile-verified
//
#include <hip/hip_runtime.h>
#include <math.h>

typedef __attribute__((ext_vector_type(2))) float v2f;
typedef __attribute__((ext_vector_type(8))) float v8f;

#define NSITES 16
#define DIM 256
#define TILES_PER_BLOCK 32
#define CHAINS_PER_WAVE 4   // 8 waves * 4 chains = 32 tiles per block

// Cyclic Jacobi eigensolve of a small symmetric PSD matrix held in registers
// (p,q,k loops fully unrolled -> static indexing), then the reference's
// entropy: p_i = lam_i / (sum + 1e-8); S = -sum p_i * log(p_i + 1e-10).
template <int N>
__device__ __forceinline__ float sym_entropy(float A[N][N]) {
  for (int sweep = 0; sweep < 6; ++sweep) {
#pragma unroll
    for (int p = 0; p < N - 1; ++p) {
#pragma unroll
      for (int q = p + 1; q < N; ++q) {
        float apq = A[p][q];
        if (fabsf(apq) > 1e-20f) {
          float tau = (A[q][q] - A[p][p]) / (2.0f * apq);
          float t = (tau >= 0.0f ? 1.0f : -1.0f) /
                    (fabsf(tau) + sqrtf(1.0f + tau * tau));
          float c = 1.0f / sqrtf(1.0f + t * t);
          float s = t * c;
#pragma unroll
          for (int k = 0; k < N; ++k) {  // A = A * J
            float akp = A[k][p], akq = A[k][q];
            A[k][p] = c * akp - s * akq;
            A[k][q] = s * akp + c * akq;
          }
#pragma unroll
          for (int k = 0; k < N; ++k) {  // A = J^T * A
            float apk = A[p][k], aqk = A[q][k];
            A[p][k] = c * apk - s * aqk;
            A[q][k] = s * apk + c * aqk;
          }
        }
      }
    }
  }
  float sum = 1e-8f;
  float lam[N];
#pragma unroll
  for (int i = 0; i < N; ++i) {
    lam[i] = fmaxf(A[i][i], 0.0f);  // eig of C*C^T == singular values squared
    sum += lam[i];
  }
  float inv = 1.0f / sum;
  float S = 0.0f;
#pragma unroll
  for (int i = 0; i < N; ++i) {
    float pn = lam[i] * inv;
    S -= pn * logf(pn + 1e-10f);
  }
  return S;
}

__global__ __launch_bounds__(256) void phi_q_kernel(
    const float* __restrict__ sites, float* __restrict__ out, int batch) {
  // 17x17 tile: stride 289 dwords (odd mod 64) -> conflict-free lane-strided
  // access in phase 2. 32 tiles = ~37 KB LDS.
  __shared__ float Cmat[TILES_PER_BLOCK][17][17];
  __shared__ float Ent[3][TILES_PER_BLOCK];

  const int lane = threadIdx.x & 31;
  const int w = threadIdx.x >> 5;

  // ---------------- Phase 1: WMMA correlation tiles ----------------
  {
    const int m = lane & 15;
    const int koff = (lane >> 4) << 1;  // K sub-offset: 0 (lanes 0-15) / 2 (16-31)
    // A rows:    0-7 -> sites 0-7 (whole A), 8-11 -> sites 0-3 (left A),
    //            12-15 -> sites 8-11 (right A)
    const int rowA = (m < 8) ? m : ((m < 12) ? (m - 8) : (m - 4));
    // B columns: 0-7 -> sites 8-15 (whole B), 8-11 -> sites 4-7 (left B),
    //            12-15 -> sites 12-15 (right B)
    const int colB = (m < 8) ? (m + 8) : ((m < 12) ? (m - 4) : m);
    const size_t sstride = (size_t)batch * DIM;
    const int rowoff = (lane >> 4) << 3;

    for (int i = 0; i < CHAINS_PER_WAVE; ++i) {
      const int t = w * CHAINS_PER_WAVE + i;
      const int b = blockIdx.x * TILES_PER_BLOCK + t;
      if (b < batch) {  // wave-uniform: EXEC all-ones for WMMA
        const float* aBase =
            sites + (size_t)rowA * sstride + (size_t)b * DIM + koff;
        const float* bBase =
            sites + (size_t)colB * sstride + (size_t)b * DIM + koff;

        v8f acc = {0.f, 0.f, 0.f, 0.f, 0.f, 0.f, 0.f, 0.f};
#pragma unroll 4
        for (int kk = 0; kk < DIM / 4; ++kk) {
          v2f a = *(const v2f*)(aBase + kk * 4);   // A[m][k..k+1] contiguous
          v2f bb = *(const v2f*)(bBase + kk * 4);  // B[k..k+1][n] contiguous
          acc = __builtin_amdgcn_wmma_f32_16x16x4_f32(
              false, a, false, bb, (short)0, acc, false, false);
        }
        // D layout: VGPR r, lanes 0-15 -> (M=r, N=lane); 16-31 -> (M=r+8, N-16)
#pragma unroll
        for (int r = 0; r < 8; ++r) Cmat[t][r + rowoff][m] = acc[r];
      }
    }
  }

  __syncthreads();

  // ---------------- Phase 2: lane-parallel Gram + Jacobi + entropy --------
  // wave 0: one 8x8 "whole" problem per lane; wave 1: left 4x4; wave 2: right 4x4.
  {
    const int t = lane;
    const int b = blockIdx.x * TILES_PER_BLOCK + t;
    if (b < batch) {
      if (w == 0) {
        float G[8][8];
#pragma unroll
        for (int i = 0; i < 8; ++i) {
#pragma unroll
          for (int j = 0; j <= i; ++j) {
            float a = 0.0f;
#pragma unroll
            for (int k = 0; k < 8; ++k) a += Cmat[t][i][k] * Cmat[t][j][k];
            G[i][j] = a;
            G[j][i] = a;
          }
        }
        Ent[0][t] = sym_entropy<8>(G);
      } else if (w == 1) {
        float G[4][4];
#pragma unroll
        for (int i = 0; i < 4; ++i) {
#pragma unroll
          for (int j = 0; j <= i; ++j) {
            float a = 0.0f;
#pragma unroll
            for (int k = 0; k < 4; ++k)
              a += Cmat[t][8 + i][8 + k] * Cmat[t][8 + j][8 + k];
            G[i][j] = a;
            G[j][i] = a;
          }
        }
        Ent[1][t] = sym_entropy<4>(G);
      } else if (w == 2) {
        float G[4][4];
#pragma unroll
        for (int i = 0; i < 4; ++i) {
#pragma unroll
          for (int j = 0; j <= i; ++j) {
            float a = 0.0f;
#pragma unroll
            for (int k = 0; k < 4; ++k)
              a += Cmat[t][12 + i][12 + k] * Cmat[t][12 + j][12 + k];
            G[i][j] = a;
            G[j][i] = a;
          }
        }
        Ent[2][t] = sym_entropy<4>(G);
      }
    }
  }

  __syncthreads();

  // ---------------- Phase 3: combine and store ----------------
  if (w == 0) {
    const int b = blockIdx.x * TILES_PER_BLOCK + lane;
    if (b < batch) {
      float phi = Ent[0][lane] - (Ent[1][lane] + Ent[2][lane]);
      out[b] = phi > 0.0f ? phi : 0.0f;
    }
  }
}

extern "C" void kernel_launch(void* const* d_in, const int* in_sizes, int n_in,
                              void* d_out, int out_size, void* d_ws,
                              size_t ws_size, hipStream_t stream) {
  const float* sites = (const float*)d_in[0];
  float* out = (float*)d_out;
  const int batch = in_sizes[0] / (NSITES * DIM);  // 8192
  const int blocks = (batch + TILES_PER_BLOCK - 1) / TILES_PER_BLOCK;  // 256
  phi_q_kernel<<<blocks, 256, 0, stream>>>(sites, out, batch);
}